// OpenLSTM_78469052498679
// MI455X (gfx1250) — compile-verified
//
#include <hip/hip_runtime.h>

typedef __attribute__((ext_vector_type(2))) float v2f;
typedef __attribute__((ext_vector_type(8))) float v8f;

#define BATCH   4096
#define TSTEPS  2048
// u_train: (B, T, 4) = [x0, x1, y0, y1] ; out: (B, T, 2)

// sigmoid(x + b) = 0.5 * tanh(0.5*x + 0.5*b) + 0.5   (hb = 0.5*b precomputed)
__device__ __forceinline__ float fsigm_b(float x, float hb) {
    return fmaf(__builtin_amdgcn_tanhf(fmaf(x, 0.5f, hb)), 0.5f, 0.5f);
}

__global__ __launch_bounds__(32)
void OpenLSTM_wmma_scan(const float* __restrict__ u,    // (B,T,4)
                        const float* __restrict__ Wih,  // (64,2)
                        const float* __restrict__ Whh,  // (64,2)
                        const float* __restrict__ bih,  // (64)
                        const float* __restrict__ bhh,  // (64)
                        const float* __restrict__ Whr,  // (2,16)
                        const int*   __restrict__ ncp,  // scalar
                        float*       __restrict__ out)  // (B,T,2)
{
    const int lane = threadIdx.x;     // 0..31
    const int n    = lane & 15;       // N index (batch-in-tile)
    const int hi   = lane >> 4;       // 0: lanes 0-15 (K0/K1), 1: lanes 16-31 (K2/K3)
    const int b0   = blockIdx.x * 16; // 16 batches per wave
    const int nc   = *ncp;            // n_context (uniform scalar load)

    // ---- A matrices (one per gate group): 16(gate) x 4(K) f32, loop-invariant.
    // A layout: lanes 0-15 hold M=0..15 {VGPR0=K0, VGPR1=K1};
    //           lanes 16-31 hold M=0..15 {VGPR0=K2, VGPR1=K3}.
    // K = [x0, x1, h0, h1]  ->  K0/K1 from W_ih, K2/K3 from W_hh.
    v2f A[4];
    const float* Wsrc = hi ? Whh : Wih;
#pragma unroll
    for (int g = 0; g < 4; ++g) {
        const int row = g * 16 + n;          // global gate row 0..63
        A[g].x = Wsrc[row * 2 + 0];
        A[g].y = Wsrc[row * 2 + 1];
    }

    // ---- biases folded into the activations (C operand is inline 0).
    // D layout: VGPR r -> M(gate row in group) = r + hi*8, N = lane&15.
    // groups: 0=i, 1=f, 2=g(tanh), 3=o
    float hbi[8], hbf[8], hbo[8], bg[8];     // hb* = 0.5*bias ; bg = full bias
#pragma unroll
    for (int r = 0; r < 8; ++r) {
        const int m = r + hi * 8;
        hbi[r] = 0.5f * (bih[ 0 + m] + bhh[ 0 + m]);
        hbf[r] = 0.5f * (bih[16 + m] + bhh[16 + m]);
        bg[r]  =         bih[32 + m] + bhh[32 + m];
        hbo[r] = 0.5f * (bih[48 + m] + bhh[48 + m]);
    }

    // ---- W_hr coefficients for the in-lane projection reduction (M = r + hi*8).
    float wr0[8], wr1[8];
#pragma unroll
    for (int r = 0; r < 8; ++r) {
        wr0[r] = Whr[0 * 16 + r + hi * 8];
        wr1[r] = Whr[1 * 16 + r + hi * 8];
    }

    // Cell state in D layout: VGPR r, lane l : c[hidden = r + hi*8][batch = b0 + (l&15)]
    v8f c = {};
    const v8f zc = {};                // WMMA C operand: inline zero
    float h0 = 0.0f, h1 = 0.0f;       // recurrent h for batch b0+n (valid in all lanes)

    // lanes 0-15 read u[b][t][0:2] (x), lanes 16-31 read u[b][t][2:4] (injected h)
    const float* up = u + (size_t)(b0 + n) * TSTEPS * 4 + hi * 2;
    float*       op = out + (size_t)(b0 + n) * TSTEPS * 2;

    // Software pipeline: ld always holds this step's 8B of input.
    float2 ld = *(const float2*)up;

    for (int t = 0; t < TSTEPS; ++t) {
        // ---- Build B (4 x 16 batches): VGPR0 = {x0 | h0}, VGPR1 = {x1 | h1}
        const bool use_mem = (!hi) || (t < nc);   // 2 v_cndmask, no exec branch
        v2f Bv;
        Bv.x = use_mem ? ld.x : h0;
        Bv.y = use_mem ? ld.y : h1;

        // issue next step's load now; waited a full compute-body later
        const int tn = (t + 1 < TSTEPS) ? (t + 1) : (TSTEPS - 1);
        const float2 ldn = *(const float2*)(up + (size_t)tn * 4);
        // +128B from the same address -> folds into the load's VGPR pair, offset:128
        __builtin_prefetch(up + (size_t)tn * 4 + 32, 0, 3);

        // ---- gates(gate x batch) = A(16x4) x B(4x16) : 4 WMMAs, C = inline 0
        v8f gi = __builtin_amdgcn_wmma_f32_16x16x4_f32(false, A[0], false, Bv, (short)0, zc, false, false);
        v8f gf = __builtin_amdgcn_wmma_f32_16x16x4_f32(false, A[1], false, Bv, (short)0, zc, false, false);
        v8f gg = __builtin_amdgcn_wmma_f32_16x16x4_f32(false, A[2], false, Bv, (short)0, zc, false, false);
        v8f go = __builtin_amdgcn_wmma_f32_16x16x4_f32(false, A[3], false, Bv, (short)0, zc, false, false);

        // ---- activations (v_tanh_f32) + cell update + in-lane projection half
        float p0 = 0.0f, p1 = 0.0f;
#pragma unroll
        for (int r = 0; r < 8; ++r) {
            const float si = fsigm_b(gi[r], hbi[r]);
            const float sf = fsigm_b(gf[r], hbf[r]);
            const float tg = __builtin_amdgcn_tanhf(gg[r] + bg[r]);
            const float so = fsigm_b(go[r], hbo[r]);
            const float cn = sf * c[r] + si * tg;
            c[r] = cn;
            const float s = so * __builtin_amdgcn_tanhf(cn);
            p0 += s * wr0[r];
            p1 += s * wr1[r];
        }
        // combine lane-halves (hidden 0-7 + hidden 8-15): every lane gets h[b0+n][:]
        h0 = p0 + __shfl_xor(p0, 16, 32);
        h1 = p1 + __shfl_xor(p1, 16, 32);

        if (!hi) {                      // one 8B non-temporal store per batch per step
            float2 hv; hv.x = h0; hv.y = h1;
            __builtin_nontemporal_store(__builtin_bit_cast(double, hv),
                                        (double*)(op + (size_t)t * 2));
        }
        ld = ldn;                       // rotate the pipeline
    }
}

extern "C" void kernel_launch(void* const* d_in, const int* in_sizes, int n_in,
                              void* d_out, int out_size, void* d_ws, size_t ws_size,
                              hipStream_t stream) {
    const float* u    = (const float*)d_in[0];
    const float* Wih  = (const float*)d_in[1];
    const float* Whh  = (const float*)d_in[2];
    const float* bih  = (const float*)d_in[3];
    const float* bhh  = (const float*)d_in[4];
    const float* Whr  = (const float*)d_in[5];
    const int*   ncp  = (const int*)d_in[6];
    float*       out  = (float*)d_out;

    dim3 grid(BATCH / 16);   // 256 single-wave workgroups -> spread across WGPs
    dim3 block(32);          // one wave32 per block, 16 batches per wave
    OpenLSTM_wmma_scan<<<grid, block, 0, stream>>>(u, Wih, Whh, bih, bhh, Whr, ncp, out);
}